// Calculate_renyi_entropy_49598282334812
// MI455X (gfx1250) — compile-verified
//
#include <hip/hip_runtime.h>
#include <cstdint>

#define BINS 50
#define HWAVES 8      // 256 threads / 32 (wave32)
#define PSTRIDE 64    // per-block histogram partial stride (channel spread)

// ---------- float <-> order-preserving uint key ----------
__device__ __forceinline__ unsigned f2key(float f) {
    unsigned u = __float_as_uint(f);
    return (u & 0x80000000u) ? ~u : (u | 0x80000000u);
}
__device__ __forceinline__ float key2f(unsigned k) {
    unsigned u = (k & 0x80000000u) ? (k & 0x7fffffffu) : ~k;
    return __uint_as_float(u);
}
__device__ __forceinline__ unsigned uminu(unsigned a, unsigned b) { return a < b ? a : b; }
__device__ __forceinline__ unsigned umaxu(unsigned a, unsigned b) { return a > b ? a : b; }

// ================= pass 1: per-block min/max partials (no global atomics) ==========
__global__ __launch_bounds__(256) void k_minmax(const float* __restrict__ x,
                                                long long n, unsigned* __restrict__ mmpart) {
    long long num4   = n >> 2;
    long long gtid   = (long long)blockIdx.x * blockDim.x + threadIdx.x;
    long long stride = (long long)gridDim.x * blockDim.x;
    float mn = 3.402823466e38f, mx = -3.402823466e38f;
    const float4* x4 = (const float4*)x;
    for (long long i = gtid; i < num4; i += stride) {
        float4 v = x4[i];
        mn = fminf(mn, fminf(fminf(v.x, v.y), fminf(v.z, v.w)));
        mx = fmaxf(mx, fmaxf(fmaxf(v.x, v.y), fmaxf(v.z, v.w)));
    }
    if (blockIdx.x == 0 && threadIdx.x < (int)(n & 3)) {     // scalar tail
        float v = x[num4 * 4 + threadIdx.x];
        mn = fminf(mn, v); mx = fmaxf(mx, v);
    }
    for (int off = 16; off > 0; off >>= 1) {                 // wave32 tree
        mn = fminf(mn, __shfl_xor(mn, off, 32));
        mx = fmaxf(mx, __shfl_xor(mx, off, 32));
    }
    __shared__ unsigned s_mn[HWAVES], s_mx[HWAVES];
    const int wv = threadIdx.x >> 5, lane = threadIdx.x & 31;
    if (lane == 0) { s_mn[wv] = f2key(mn); s_mx[wv] = f2key(mx); }
    __syncthreads();
    if (threadIdx.x == 0) {
        unsigned a = s_mn[0], b = s_mx[0];
        for (int w = 1; w < HWAVES; w++) { a = uminu(a, s_mn[w]); b = umaxu(b, s_mx[w]); }
        mmpart[2 * blockIdx.x]     = a;   // plain stores, distinct addresses
        mmpart[2 * blockIdx.x + 1] = b;
    }
}

// ================= reduce min/max partials -> ws[0], ws[1] =========================
__global__ __launch_bounds__(256) void k_mmreduce(const unsigned* __restrict__ mmpart,
                                                  int nblk, unsigned* __restrict__ ws) {
    unsigned a = 0xffffffffu, b = 0u;
    for (int i = threadIdx.x; i < nblk; i += blockDim.x) {
        a = uminu(a, mmpart[2 * i]);
        b = umaxu(b, mmpart[2 * i + 1]);
    }
    for (int off = 16; off > 0; off >>= 1) {
        a = uminu(a, (unsigned)__shfl_xor((int)a, off, 32));
        b = umaxu(b, (unsigned)__shfl_xor((int)b, off, 32));
    }
    __shared__ unsigned sa[HWAVES], sb[HWAVES];
    const int wv = threadIdx.x >> 5, lane = threadIdx.x & 31;
    if (lane == 0) { sa[wv] = a; sb[wv] = b; }
    __syncthreads();
    if (threadIdx.x == 0) {
        for (int w = 1; w < HWAVES; w++) { a = uminu(a, sa[w]); b = umaxu(b, sb[w]); }
        ws[0] = a; ws[1] = b;
    }
}

// ================= pass 2: histogram ===============================================
// - global->LDS staging via async-tensor path (double buffered, ASYNCcnt)
// - per-lane privatized counters: s_cnt[wave][bin][lane]; bank = (bin*32+lane)%64
//   => lanes always hit 32 distinct banks; ds_add has zero same-address serialization
__global__ __launch_bounds__(256) void k_hist(const float* __restrict__ x,
                                              long long n,
                                              const unsigned* __restrict__ ws,
                                              unsigned* __restrict__ hpart) {
    __shared__ float4   s_stage[HWAVES][2][32];        // 8 KB staging
    __shared__ unsigned s_cnt[HWAVES][BINS][32];       // 50 KB privatized counters

    const int lane = threadIdx.x & 31;
    const int wv   = threadIdx.x >> 5;

    // each lane zeroes its own private column (conflict-free, no barrier needed)
    unsigned* __restrict__ hp = &s_cnt[wv][0][lane];
    #pragma unroll
    for (int b = 0; b < BINS; b++) hp[b * 32] = 0u;

    const float mn    = key2f(ws[0]);
    const float mx    = key2f(ws[1]);
    const float width = fmaxf(mx - mn, 1e-12f);
    const float scale = (float)BINS / width;
    const float c0    = -mn * scale;

    const long long tiles = n >> 7;                    // 128 floats per wave-tile
    const long long nw    = ((long long)gridDim.x * blockDim.x) >> 5;
    long long t           = ((long long)blockIdx.x * blockDim.x + threadIdx.x) >> 5;

    const unsigned lds0 = (unsigned)(uintptr_t)&s_stage[wv][0][lane];
    const unsigned lds1 = (unsigned)(uintptr_t)&s_stage[wv][1][lane];
    const char* base = (const char*)x;

    int buf = 0;
    if (t < tiles) {
        const char* ga = base + (t << 9) + ((long long)lane << 4);
        asm volatile("global_load_async_to_lds_b128 %0, %1, off"
                     :: "v"(lds0), "v"(ga) : "memory");
    }
    while (t < tiles) {
        long long tn = t + nw;
        unsigned ldsN = buf ? lds0 : lds1;
        if (tn < tiles) {                              // prefetch next tile
            const char* ga = base + (tn << 9) + ((long long)lane << 4);
            asm volatile("global_load_async_to_lds_b128 %0, %1, off"
                         :: "v"(ldsN), "v"(ga) : "memory");
            asm volatile("s_wait_asynccnt 0x1" ::: "memory");
        } else {
            asm volatile("s_wait_asynccnt 0x0" ::: "memory");
        }
        float4 v = s_stage[wv][buf][lane];             // ds_load_b128
        #pragma unroll
        for (int c = 0; c < 4; c++) {
            float e = (c == 0) ? v.x : (c == 1) ? v.y : (c == 2) ? v.z : v.w;
            int b = (int)floorf(fmaf(e, scale, c0));
            b = b < 0 ? 0 : (b > BINS - 1 ? BINS - 1 : b);
            atomicAdd(&hp[b * 32], 1u);                // ds_add_u32, private slot
        }
        t = tn; buf ^= 1;
    }

    // scalar tail into this thread's own private slot (before merge barrier)
    long long done = tiles << 7;
    if (blockIdx.x == 0 && threadIdx.x < (int)(n - done)) {
        float e = x[done + threadIdx.x];
        int b = (int)floorf(fmaf(e, scale, c0));
        b = b < 0 ? 0 : (b > BINS - 1 ? BINS - 1 : b);
        atomicAdd(&hp[b * 32], 1u);
    }
    __syncthreads();

    // block merge: 50 threads each fold 8 waves x 32 lanes (uint4 loads), plain store
    for (int b = threadIdx.x; b < BINS; b += blockDim.x) {
        unsigned s = 0;
        for (int w = 0; w < HWAVES; w++) {
            const uint4* row = (const uint4*)&s_cnt[w][b][0];
            #pragma unroll
            for (int q = 0; q < 8; q++) {
                uint4 u = row[q];
                s += u.x + u.y + u.z + u.w;
            }
        }
        hpart[(size_t)blockIdx.x * PSTRIDE + b] = s;
    }
}

// ======= reduce per-block histogram partials: block b sums bin b -> ws[2+b] ========
__global__ __launch_bounds__(256) void k_histreduce(const unsigned* __restrict__ hpart,
                                                    int nblk, unsigned* __restrict__ ws) {
    const int b = blockIdx.x;                          // 0..BINS-1
    unsigned s = 0;
    for (int i = threadIdx.x; i < nblk; i += blockDim.x)
        s += hpart[(size_t)i * PSTRIDE + b];
    for (int off = 16; off > 0; off >>= 1)
        s += (unsigned)__shfl_xor((int)s, off, 32);
    __shared__ unsigned sw[HWAVES];
    const int wv = threadIdx.x >> 5, lane = threadIdx.x & 31;
    if (lane == 0) sw[wv] = s;
    __syncthreads();
    if (threadIdx.x == 0) {
        for (int w = 1; w < HWAVES; w++) s += sw[w];
        ws[2 + b] = s;
    }
}

// ================= finalize: -log2(sum p^2) ========================================
__global__ void k_final(const unsigned* __restrict__ ws, float* __restrict__ out) {
    if (threadIdx.x == 0 && blockIdx.x == 0) {
        float total = 0.f;
        for (int i = 0; i < BINS; i++) total += (float)ws[2 + i];
        float s = 0.f;
        for (int i = 0; i < BINS; i++) {
            float p = (float)ws[2 + i] / total;
            s += p * p;
        }
        out[0] = -log2f(s);                            // 1/(1-ALPHA) * log2, ALPHA=2
    }
}

extern "C" void kernel_launch(void* const* d_in, const int* in_sizes, int n_in,
                              void* d_out, int out_size, void* d_ws, size_t ws_size,
                              hipStream_t stream) {
    const float* x = (const float*)d_in[0];
    long long total = (long long)in_sizes[0];
    long long n = total / 4;              // only time_freq_map[0] of (4,4096,8192)
    unsigned* wsu = (unsigned*)d_ws;

    // ws layout (u32 words): [0..1] min/max keys, [2..51] final bins,
    // [64 ..) min/max partials (2*NB1), then histogram partials (PSTRIDE*NB2)
    long long words = (long long)(ws_size / 4);
    int NB1 = 2048;
    if (64 + 2LL * NB1 + PSTRIDE > words) {
        NB1 = (int)((words - 64 - PSTRIDE) / 2);
        if (NB1 < 1) NB1 = 1;
    }
    long long HP = 64 + 2LL * NB1;
    int NB2 = 2048;
    long long cap = (words - HP) / PSTRIDE;
    if (cap < NB2) NB2 = (int)(cap < 1 ? 1 : cap);

    unsigned* mmpart = wsu + 64;
    unsigned* hpart  = wsu + HP;

    k_minmax    <<<NB1, 256, 0, stream>>>(x, n, mmpart);
    k_mmreduce  <<<1,   256, 0, stream>>>(mmpart, NB1, wsu);
    k_hist      <<<NB2, 256, 0, stream>>>(x, n, wsu, hpart);
    k_histreduce<<<BINS,256, 0, stream>>>(hpart, NB2, wsu);
    k_final     <<<1,   32,  0, stream>>>(wsu, (float*)d_out);
}